// WeightedLagDenseLoss_21345987461387
// MI455X (gfx1250) — compile-verified
//
#include <hip/hip_runtime.h>

// CDNA5 / gfx1250 — wave32, WMMA f32 16x16x4, LDS-staged B-operand panel.
//
// total[b,t] = (p-g)^2 + 0.5*( H*p^2 - 2*p*S1[b,t] + S2[b,t] ), weighted, meaned.
// S1 = tgt   @ M^T   (128x512)x(512x512)     S2 = tgt^2 @ M^T
// M[t][j] = 1/(j-t) for t<j<T-1 ; M[t][T-1] = C[t] = sum_{l=max(1,T-1-t)}^{T-1} 1/l.
//
// Block layout: one block per t-tile (16 output columns). Its 8 waves share the
// same B-operand (coefficient) panel and cover the 8 b-tiles. The panel is
// built once into LDS in WMMA B-operand order: panel[step][lane] (float2), so
// the hot loop's LDS read is a conflict-free ds_load_b64 and contains no
// branches (EXEC must be all-ones around WMMA).

typedef float v2f __attribute__((ext_vector_type(2)));
typedef float v8f __attribute__((ext_vector_type(8)));

#define LAG_PENALTY 0.5f
#define T_DIM 512
#define KSTEPS (T_DIM / 4)   // 128 WMMA k-steps

__global__ void zero_out_kernel(float* out) { out[0] = 0.0f; }

__global__ __launch_bounds__(256) void lagloss_wmma_kernel(
    const float* __restrict__ pred,
    const float* __restrict__ tgt,
    const float* __restrict__ wgt,
    float* __restrict__ out,
    int B, int T)
{
    // B-operand panel in WMMA lane order: entry (s, lane) = float2 holding
    // B[k= 2*khi + {0,1}][n=nl] for k-step s, i.e. M[t0+nl][4s + 2*khi + {0,1}].
    __shared__ float panel[KSTEPS * 32 * 2];   // 32 KB

    const int tid  = threadIdx.x;
    const int lane = tid & 31;
    const int wave = tid >> 5;                 // 8 waves = 8 b-tiles
    const int t0   = blockIdx.x << 4;          // this block's t-tile
    const int b0   = wave << 4;                // this wave's b-tile
    const int Tm1  = T - 1;

    // ---- Cooperative panel fill (branchy part lives here, far from WMMA) ----
    for (int e = tid; e < KSTEPS * 32; e += 256) {
        const int s   = e >> 5;
        const int L   = e & 31;
        const int nl  = L & 15;
        const int khi = L >> 4;
        const int t   = t0 + nl;
        const int j0  = 4 * s + 2 * khi;

        float c0 = (j0     > t) ? __builtin_amdgcn_rcpf((float)(j0     - t)) : 0.0f;
        float c1 = (j0 + 1 > t) ? __builtin_amdgcn_rcpf((float)(j0 + 1 - t)) : 0.0f;
        panel[e * 2 + 0] = c0;
        panel[e * 2 + 1] = c1;
    }
    __syncthreads();

    // Boundary column j == T-1 carries C[t] = sum_{l=max(1,T-1-t)}^{T-1} 1/l.
    // That entry is (s = T/4-1, khi = 1, element 1). 16 threads patch it.
    if (tid < 16) {
        const int t = t0 + tid;
        int lstart = Tm1 - t; if (lstart < 1) lstart = 1;
        float Ct = 0.0f;
        for (int l = lstart; l <= Tm1; ++l)
            Ct += __builtin_amdgcn_rcpf((float)l);
        const int s = (T >> 2) - 1;
        panel[((s * 32) + 16 + tid) * 2 + 1] = Ct;
    }
    __syncthreads();

    // H = H_{T-1} = sum_{l=1}^{T-1} 1/l  (uniform trip count, no divergence)
    float H = 0.0f;
    for (int l = 1; l <= Tm1; ++l)
        H += __builtin_amdgcn_rcpf((float)l);

    // ---- GEMM main loop ----
    // A-matrix 16x4 f32 layout: lane m = lane&15, K = 2*(lane>>4) + vgpr.
    const int nl   = lane & 15;
    const int khi  = lane >> 4;
    const int t    = t0 + nl;                  // this lane's output column
    const int arow = (b0 + nl) * T;

    v8f c1 = {0.f,0.f,0.f,0.f,0.f,0.f,0.f,0.f};   // S1 tile
    v8f c2 = {0.f,0.f,0.f,0.f,0.f,0.f,0.f,0.f};   // S2 tile

    const float* aptr = tgt + arow + 2 * khi;
    const float* bptr = panel + lane * 2;      // conflict-free: dwords 2L, 2L+1

    #pragma unroll 4
    for (int s = 0; s < KSTEPS; ++s) {
        v2f a  = *(const v2f*)(aptr + 4 * s);          // global_load_b64
        v2f bb = *(const v2f*)(bptr + 64 * s);         // ds_load_b64
        v2f a2 = a * a;                                // v_pk_mul_f32

        c1 = __builtin_amdgcn_wmma_f32_16x16x4_f32(
                 false, a,  false, bb, (short)0, c1, false, false);
        c2 = __builtin_amdgcn_wmma_f32_16x16x4_f32(
                 false, a2, false, bb, (short)0, c2, false, false);
    }

    // ---- Fused epilogue ----
    // C/D layout: VGPR v, lane l -> M = v + 8*(l>=16), N = l&15.
    float acc = 0.0f;
    #pragma unroll
    for (int v = 0; v < 8; ++v) {
        const int   b  = b0 + v + 8 * khi;
        const float p  = pred[b * T + t];
        const float g  = tgt [b * T + t];
        const float s1 = c1[v];
        const float s2 = c2[v];
        const float lagl = H * p * p - 2.0f * p * s1 + s2;
        const float d  = p - g;
        acc += wgt[b] * (d * d + LAG_PENALTY * lagl);
    }
    acc *= 1.0f / (float)(B * T);              // mean over B*T

    // wave32 butterfly reduction, one atomic per wave.
    for (int off = 16; off > 0; off >>= 1)
        acc += __shfl_xor(acc, off, 32);
    if (lane == 0)
        atomicAdd(out, acc);
}

extern "C" void kernel_launch(void* const* d_in, const int* in_sizes, int n_in,
                              void* d_out, int out_size, void* d_ws, size_t ws_size,
                              hipStream_t stream) {
    const float* pred = (const float*)d_in[0];
    const float* tgt  = (const float*)d_in[1];
    const float* wgt  = (const float*)d_in[2];
    float* out = (float*)d_out;

    const int B = in_sizes[2];          // 128
    const int T = in_sizes[0] / B;      // 512

    // d_out is poisoned by the harness and accumulated via atomics -> zero it
    // first (stream-ordered before the main kernel).
    zero_out_kernel<<<1, 1, 0, stream>>>(out);

    const int tilesT = T / 16;          // 32 blocks, 8 waves (b-tiles) each
    lagloss_wmma_kernel<<<tilesT, 256, 0, stream>>>(pred, tgt, wgt, out, B, T);
}